// SimpleRiskGNN_90537910600316
// MI455X (gfx1250) — compile-verified
//
#include <hip/hip_runtime.h>
#include <hip/hip_bf16.h>
#include <math.h>

typedef __attribute__((ext_vector_type(2))) float v2f;
typedef __attribute__((ext_vector_type(8))) float v8f;

#define F_DIM 64

// ---------------------------------------------------------------------------
// Phase 0: zero the accumulator / degree scratch arrays (must be done every
// call: the harness only poisons once and replays the graph).
// ---------------------------------------------------------------------------
__global__ void zero_kernel(float* __restrict__ p, int n) {
    int i = blockIdx.x * blockDim.x + threadIdx.x;
    if (i < n) p[i] = 0.0f;
}

// ---------------------------------------------------------------------------
// Phase 1: s = X @ W  (per-node scalar), done with V_WMMA_F32_16X16X4_F32.
// One wave32 handles a 16-row tile of X. A = 16x4 f32 slab of X, B = 4x16
// with only column N=0 populated (W), C accumulates 16x16 but only column 0
// is meaningful. K loops 0..63 in steps of 4 -> 16 chained WMMAs.
//
// ISA VGPR layouts used:
//   A 16x4 f32 : lanes 0-15 (M=lane)   v0=K0, v1=K1 ; lanes 16-31 v0=K2, v1=K3
//   B 4x16 f32 : lanes 0-15 (N=lane)   v0=K0, v1=K1 ; lanes 16-31 v0=K2, v1=K3
//   C 16x16 f32: lane n (n<16) vr = C[M=r][n]; lane n+16 vr = C[M=r+8][n]
// => column 0 of C lives in lanes 0 (M=0..7) and 16 (M=8..15).
// ---------------------------------------------------------------------------
__global__ void __launch_bounds__(64)
matvec_wmma_kernel(const float* __restrict__ X, const float* __restrict__ W,
                   float* __restrict__ s, int n_nodes) {
    const int lane = threadIdx.x & 31;
    const int wave = threadIdx.x >> 5;
    const int tile = blockIdx.x * (blockDim.x >> 5) + wave;
    const int base = tile << 4;               // first row of this 16-row tile
    if (base >= n_nodes) return;              // uniform per wave

    const int m    = lane & 15;               // M index (A) / N index (B)
    const int half = lane >> 4;               // selects K-pair {0,1} vs {2,3}

    const float* __restrict__ xrow = X + (size_t)(base + m) * F_DIM;

    v8f c = {};                               // f32 accumulator (column 0 used)
#pragma unroll
    for (int k = 0; k < F_DIM; k += 4) {
        // A fragment: this lane's row, K = k + 2*half .. k + 2*half + 1
        const v2f a = *reinterpret_cast<const v2f*>(xrow + k + (half << 1));
        // B fragment: only N==0 lanes (m==0) carry W; everything else is 0
        const v2f wv = *reinterpret_cast<const v2f*>(W + k + (half << 1));
        v2f bf;
        bf.x = (m == 0) ? wv.x : 0.0f;
        bf.y = (m == 0) ? wv.y : 0.0f;
        // 8 args: (neg_a, A, neg_b, B, c_mod, C, reuse_a, reuse_b)
        c = __builtin_amdgcn_wmma_f32_16x16x4_f32(false, a, false, bf,
                                                  (short)0, c, false, false);
    }

    // Column N=0 of C: lane 0 holds M=0..7, lane 16 holds M=8..15.
    if (m == 0) {
        float* __restrict__ sp = s + base + (half << 3);
#pragma unroll
        for (int r = 0; r < 8; ++r) sp[r] = c[r];
    }
}

// ---------------------------------------------------------------------------
// Phase 2: per-edge scalar scatter:  acc[dst] += s[src];  deg[dst] += 1
// s/acc/deg are 400 KB each -> resident in L2; atomics resolve in L2.
// ---------------------------------------------------------------------------
__global__ void __launch_bounds__(256)
edge_scatter_kernel(const int* __restrict__ src, const int* __restrict__ dst,
                    const float* __restrict__ s, float* __restrict__ acc,
                    float* __restrict__ deg, int n_edges) {
    int e = blockIdx.x * blockDim.x + threadIdx.x;
    if (e >= n_edges) return;
    const int si = src[e];
    const int di = dst[e];
    const float sv = s[si];
    atomicAdd(&acc[di], sv);      // non-returning -> global_atomic_add_f32
    atomicAdd(&deg[di], 1.0f);
}

// ---------------------------------------------------------------------------
// Phase 3: out = sigmoid(acc / max(deg,1) + b)
// ---------------------------------------------------------------------------
__global__ void __launch_bounds__(256)
finalize_kernel(const float* __restrict__ acc, const float* __restrict__ deg,
                const float* __restrict__ b, float* __restrict__ out, int n) {
    int i = blockIdx.x * blockDim.x + threadIdx.x;
    if (i >= n) return;
    const float d = fmaxf(deg[i], 1.0f);
    const float x = acc[i] / d + b[0];
    out[i] = 1.0f / (1.0f + expf(-x));
}

// ---------------------------------------------------------------------------
extern "C" void kernel_launch(void* const* d_in, const int* in_sizes, int n_in,
                              void* d_out, int out_size, void* d_ws, size_t ws_size,
                              hipStream_t stream) {
    const float* X     = (const float*)d_in[0];   // (N, 64) f32
    const int*   edges = (const int*)d_in[1];     // (2, E) int
    const float* W     = (const float*)d_in[2];   // (64, 1) f32
    const float* b     = (const float*)d_in[3];   // (1,)    f32
    float*       out   = (float*)d_out;           // (N,)    f32

    const int n_nodes = in_sizes[0] / F_DIM;
    const int n_edges = in_sizes[1] / 2;
    const int* src = edges;
    const int* dst = edges + n_edges;

    // Workspace layout: s | acc | deg  (n_nodes floats each)
    float* s   = (float*)d_ws;
    float* acc = s + n_nodes;
    float* deg = acc + n_nodes;

    // Phase 0: zero acc+deg (contiguous 2*n_nodes floats)
    {
        const int n = 2 * n_nodes;
        zero_kernel<<<(n + 255) / 256, 256, 0, stream>>>(acc, n);
    }

    // Phase 1: s = X @ W via f32 WMMA (one wave per 16-row tile)
    {
        const int tiles  = (n_nodes + 15) / 16;   // 6250, exact
        const int blocks = (tiles + 1) / 2;       // 2 waves per 64-thread block
        matvec_wmma_kernel<<<blocks, 64, 0, stream>>>(X, W, s, n_nodes);
    }

    // Phase 2: per-edge scalar scatter-add
    edge_scatter_kernel<<<(n_edges + 255) / 256, 256, 0, stream>>>(
        src, dst, s, acc, deg, n_edges);

    // Phase 3: mean + bias + sigmoid
    finalize_kernel<<<(n_nodes + 255) / 256, 256, 0, stream>>>(
        acc, deg, b, out, n_nodes);
}